// EpisodicCuriosity_37237366456343
// MI455X (gfx1250) — compile-verified
//
#include <hip/hip_runtime.h>
#include <math.h>
#include <stdint.h>

// Problem constants (fixed by setup_inputs)
#define T_ENVS 64
#define BQ     32     // queries per env
#define MM     4096   // memory slots per env
#define FF     512    // feature dim
#define KS     32     // K slab per iteration
#define NSLAB  (FF / KS)   // 16
#define CPB    128    // memory columns per block
#define NCB    (MM / CPB)  // 32 col-blocks
#define LDA    36     // LDS row stride for slabs (bank-conflict-free, 16B aligned)
#define LDD    132    // LDS row stride for d2 tile
#define KNN    10

typedef __attribute__((ext_vector_type(2))) float v2f;
typedef __attribute__((ext_vector_type(8))) float v8f;

// Branchless insert of v into a descending-sorted register list of length 10.
__device__ __forceinline__ void topk_insert(float (&t)[KNN], float v) {
#pragma unroll
    for (int j = 0; j < KNN; j++) {
        float hi = fmaxf(t[j], v);
        float lo = fminf(t[j], v);
        t[j] = hi;
        v = lo;
    }
}

// One 16B lane-chunk: global -> LDS direct (ASYNCcnt-tracked, no VGPR staging).
__device__ __forceinline__ void async_copy_b128(uint32_t lds_off, const float* gaddr) {
    asm volatile("global_load_async_to_lds_b128 %0, %1, off"
                 :: "v"(lds_off), "v"(gaddr)
                 : "memory");
}

__device__ __forceinline__ void wait_async0() {
    asm volatile("s_wait_asynccnt 0x0" ::: "memory");
}

// Stage one mem K-slab (128 rows x 32 floats) asynchronously into `buf`.
// chunk = i*256 + tid -> per wave-instruction: 4 rows x full 128B line.
__device__ __forceinline__ void stage_b_async(const float* gmem, int kb, float* buf, int tid) {
#pragma unroll
    for (int i = 0; i < 4; i++) {
        const int ch  = i * 256 + tid;  // 0..1023 16B chunks
        const int row = ch >> 3;        // 0..127
        const int seg = ch & 7;         // 0..7
        const float* g = gmem + (size_t)row * FF + kb + seg * 4;
        const uint32_t loff = (uint32_t)(uintptr_t)(buf + row * LDA + seg * 4);
        async_copy_b128(loff, g);
    }
}

// Stage one enc K-slab (32 rows x 32 floats) asynchronously: 1 chunk per thread,
// 4 rows x full 128B line per wave-instruction.
__device__ __forceinline__ void stage_a_async(const float* genc, int kb, float* buf,
                                              int arow, int aseg) {
    const float* g = genc + (size_t)arow * FF + kb + aseg * 4;
    const uint32_t loff = (uint32_t)(uintptr_t)(buf + arow * LDA + aseg * 4);
    async_copy_b128(loff, g);
}

__global__ __launch_bounds__(256) void ec_dist_topk_kernel(
    const float* __restrict__ enc,   // (T, B, F)
    const float* __restrict__ mem,   // (T, M, F)
    float* __restrict__ partials)    // (T, NCB, B, KNN)
{
    const int cb  = blockIdx.x;      // column block 0..31
    const int env = blockIdx.y;      // env 0..63
    const int tid  = threadIdx.x;
    const int wave = tid >> 5;       // 0..7 -> col tile
    const int lane = tid & 31;
    const int half = lane >> 4;      // 0: K pair {0,1} / M 0..7 ; 1: K pair {2,3} / M 8..15
    const int sub  = lane & 15;
    const int arow = tid >> 3;       // enc staging: row 0..31
    const int aseg = tid & 7;        // enc staging: 16B segment 0..7

    __shared__ __align__(16) float bS2[2][CPB * LDA];  // 2 x 18432 B, mem slab [col][k]
    __shared__ __align__(16) float aS2[2][BQ * LDA];   // 2 x  4608 B, enc slab [row][k]
    __shared__ __align__(16) float d2S[BQ * LDD];      // 16896 B, d2 tile [row][col]
    __shared__ float e2s[BQ];                          // ||enc row||^2
    __shared__ float m2s[CPB];                         // ||mem col||^2
    float* ltop = bS2[0];                              // alias: 2560 floats <= 4608

    if (tid < BQ)  e2s[tid] = 0.0f;
    if (tid < CPB) m2s[tid] = 0.0f;
    __syncthreads();

    const float* gmem = mem + ((size_t)env * MM + (size_t)cb * CPB) * FF;
    const float* genc = enc + (size_t)env * BQ * FF;

    // Prologue: kick off slab 0 into buffer 0 (all async, ASYNCcnt-tracked).
    stage_b_async(gmem, 0, bS2[0], tid);
    stage_a_async(genc, 0, aS2[0], arow, aseg);

    v8f acc0 = {};  // row tile 0 (queries 0..15)
    v8f acc1 = {};  // row tile 1 (queries 16..31)

    for (int s = 0; s < NSLAB; s++) {
        const int cur = s & 1;
        wait_async0();    // this wave's slab-s copies have landed in LDS
        __syncthreads();  // all waves landed; all done reading the other buffer

        // Overlap: start slab s+1 copies while computing slab s.
        if (s + 1 < NSLAB) {
            stage_b_async(gmem, (s + 1) * KS, bS2[cur ^ 1], tid);
            stage_a_async(genc, (s + 1) * KS, aS2[cur ^ 1], arow, aseg);
        }

        // ---- WMMA: wave w owns col tile w; accumulate both row tiles ----
        // A 16x4 fp32 layout: lanes 0-15 hold {K=0,1}, lanes 16-31 hold {K=2,3} (M = sub)
        // B 4x16 fp32 layout mirrors A with N = sub.
        const float* bB  = bS2[cur] + (wave * 16 + sub) * LDA;
        const float* aB0 = aS2[cur] + sub * LDA;
        const float* aB1 = aS2[cur] + (16 + sub) * LDA;
#pragma unroll
        for (int kk = 0; kk < KS; kk += 4) {
            const int ko = kk + half * 2;
            v2f b, a0, a1;
            b.x  = bB[ko];  b.y  = bB[ko + 1];
            a0.x = aB0[ko]; a0.y = aB0[ko + 1];
            a1.x = aB1[ko]; a1.y = aB1[ko + 1];
            acc0 = __builtin_amdgcn_wmma_f32_16x16x4_f32(false, a0, false, b,
                                                         (short)0, acc0, false, false);
            acc1 = __builtin_amdgcn_wmma_f32_16x16x4_f32(false, a1, false, b,
                                                         (short)0, acc1, false, false);
        }

        // ---- ||m||^2 from LDS read-back (data stable until end barrier) ----
        {
            const int brow = tid >> 1;
            const int bseg = tid & 1;
            const float* srcp = bS2[cur] + brow * LDA + bseg * 16;
            float ssq = 0.0f;
#pragma unroll
            for (int i = 0; i < 16; i++) ssq += srcp[i] * srcp[i];
            atomicAdd(&m2s[brow], ssq);  // ds_add_f32
        }
        // ---- ||e||^2 from LDS read-back ----
        {
            const float* srcp = aS2[cur] + arow * LDA + aseg * 4;
            atomicAdd(&e2s[arow], srcp[0] * srcp[0] + srcp[1] * srcp[1] +
                                  srcp[2] * srcp[2] + srcp[3] * srcp[3]);
        }
        __syncthreads();  // slab fully consumed; safe to overwrite next round
    }

    // ---- form clamped squared distances in LDS ----
    // C layout: VGPR i -> M = i + 8*half (per 16x16 f32 C map), N = sub
    const int cc = wave * 16 + sub;
#pragma unroll
    for (int i = 0; i < 8; i++) {
        const int r0 = i + half * 8;
        const int r1 = 16 + r0;
        float d0 = fmaxf(e2s[r0] + m2s[cc] - 2.0f * acc0[i], 0.0f);
        float d1 = fmaxf(e2s[r1] + m2s[cc] - 2.0f * acc1[i], 0.0f);
        d2S[r0 * LDD + cc] = d0;
        d2S[r1 * LDD + cc] = d1;
    }
    __syncthreads();

    // ---- stage 1 top-k: 8 threads/row, 16 cols each ----
    {
        const int row = tid >> 3;
        const int seg = tid & 7;
        float t10[KNN];
#pragma unroll
        for (int j = 0; j < KNN; j++) t10[j] = -1.0f;
        const float* src = d2S + row * LDD + seg * 16;
#pragma unroll
        for (int c = 0; c < 16; c++) topk_insert(t10, src[c]);
        float* dst = ltop + (row * 8 + seg) * KNN;
#pragma unroll
        for (int j = 0; j < KNN; j++) dst[j] = t10[j];
    }
    __syncthreads();

    // ---- stage 2: merge 8x10 -> top-10 per row, write partials ----
    if (tid < BQ) {
        const int row = tid;
        float t10[KNN];
#pragma unroll
        for (int j = 0; j < KNN; j++) t10[j] = -1.0f;
        const float* src = ltop + row * 8 * KNN;
        for (int c = 0; c < 8 * KNN; c++) topk_insert(t10, src[c]);
        float* dst = partials + (((size_t)env * NCB + cb) * BQ + row) * KNN;
#pragma unroll
        for (int j = 0; j < KNN; j++) dst[j] = t10[j];
    }
}

__global__ __launch_bounds__(32) void ec_merge_scan_kernel(
    const float* __restrict__ partials,  // (T, NCB, B, KNN)
    float* __restrict__ out)             // (T, B)
{
    const int env = blockIdx.x;
    const int b = threadIdx.x;  // 0..31
    __shared__ float knn[BQ][KNN];

    float t10[KNN];
#pragma unroll
    for (int j = 0; j < KNN; j++) t10[j] = -1.0f;
    for (int cb = 0; cb < NCB; cb++) {
        const float* p = partials + (((size_t)env * NCB + cb) * BQ + b) * KNN;
#pragma unroll
        for (int j = 0; j < KNN; j++) topk_insert(t10, p[j]);
    }
#pragma unroll
    for (int j = 0; j < KNN; j++) knn[b][j] = t10[j];
    __syncthreads();

    // sequential running-mean scan over the batch dim (matches jax.lax.scan)
    if (b == 0) {
        float rm[KNN];
#pragma unroll
        for (int j = 0; j < KNN; j++) rm[j] = 0.0f;
        float n = 0.0f;
        for (int i = 0; i < BQ; i++) {
            n += 1.0f;
            float s = 0.0f;
#pragma unroll
            for (int j = 0; j < KNN; j++) {
                rm[j] += (knn[i][j] - rm[j]) / n;
                float nd = fmaxf(knn[i][j] / rm[j] - 0.008f, 0.0f);
                s += 0.001f / (nd + 0.001f);
            }
            float sim = sqrtf(s) + 0.01f;
            out[(size_t)env * BQ + i] = (sim > 8.0f) ? 0.0f : (1.0f / sim);
        }
    }
}

extern "C" void kernel_launch(void* const* d_in, const int* in_sizes, int n_in,
                              void* d_out, int out_size, void* d_ws, size_t ws_size,
                              hipStream_t stream) {
    (void)in_sizes; (void)n_in; (void)out_size; (void)ws_size;
    const float* enc = (const float*)d_in[0];  // (64, 32, 512)
    const float* mem = (const float*)d_in[1];  // (64, 4096, 512)
    float* out = (float*)d_out;                // (64, 32)
    float* partials = (float*)d_ws;            // 64*32*32*10 floats = 2.5 MB

    dim3 g1(NCB, T_ENVS);
    ec_dist_topk_kernel<<<g1, 256, 0, stream>>>(enc, mem, partials);
    ec_merge_scan_kernel<<<T_ENVS, 32, 0, stream>>>(partials, out);
}